// Encoder_24970939859195
// MI455X (gfx1250) — compile-verified
//
#include <hip/hip_runtime.h>
#include <math.h>

typedef __attribute__((ext_vector_type(2))) float v2f;
typedef __attribute__((ext_vector_type(8))) float v8f;

// Tiny MLP (6->8->4->2->1, relu/relu/relu/softplus) over 4M rows.
// Memory bound (AI ~6.7 F/B): stream x once, write out once.
// Layer 1 via V_WMMA_F32_16X16X4_F32 as D = W1 * X^T (K=6 split 4+2):
//   - B operand (4x16 f32): VGPR0 = {lanes0-15: k=0, lanes16-31: k=2},
//     VGPR1 = {k=1, k=3}; column n = lane%16 = row index. This maps to
//     plain float2 loads from the row-major input, no shuffling.
//   - D (16x16 f32): lane n<16 holds feature m in VGPR m for row n,
//     i.e. each row's full hidden vector lands in ONE lane -> layers
//     7/8/9 + softplus finish per-lane with plain FMAs.
__global__ __launch_bounds__(256) void mlp6_8_4_2_1_softplus(
    const float* __restrict__ x,
    const float* __restrict__ W1, const float* __restrict__ b1,
    const float* __restrict__ W7, const float* __restrict__ b7,
    const float* __restrict__ W8, const float* __restrict__ b8,
    const float* __restrict__ W9, const float* __restrict__ b9,
    float* __restrict__ out, int n)
{
    const int lane = threadIdx.x & 31;
    const int hi   = lane >> 4;    // 0 for lanes 0-15, 1 for lanes 16-31
    const int row  = lane & 15;

    // ---- A operands: W1 tiles in WMMA A layout (16x4, M=lane%16,
    // K = {0,1} in v0/v1 for lanes<16, {2,3} for lanes>=16). Rows 8..15 zero.
    v2f a0 = {0.f, 0.f};           // covers k = 0..3
    v2f a1 = {0.f, 0.f};           // covers k = 4,5 (k=6,7 zero padding)
    if (row < 8) {
        const int c0 = hi * 2;
        a0.x = W1[row * 6 + c0];
        a0.y = W1[row * 6 + c0 + 1];
        if (hi == 0) {
            a1.x = W1[row * 6 + 4];
            a1.y = W1[row * 6 + 5];
        }
    }

    // ---- Wave-uniform small weights (compiler scalarizes these) ----
    float B1r[8], W7r[4][8], B7r[4], W8r[2][4], B8r[2], W9r[2], B9v;
#pragma unroll
    for (int j = 0; j < 8; ++j) B1r[j] = b1[j];
#pragma unroll
    for (int i = 0; i < 4; ++i) {
        B7r[i] = b7[i];
#pragma unroll
        for (int j = 0; j < 8; ++j) W7r[i][j] = W7[i * 8 + j];
    }
#pragma unroll
    for (int i = 0; i < 2; ++i) {
        B8r[i] = b8[i];
#pragma unroll
        for (int j = 0; j < 4; ++j) W8r[i][j] = W8[i * 4 + j];
    }
    W9r[0] = W9[0]; W9r[1] = W9[1]; B9v = b9[0];

    const int numTiles      = (n + 15) >> 4;
    const int wavesPerBlock = blockDim.x >> 5;
    const int waveId        = blockIdx.x * wavesPerBlock + (threadIdx.x >> 5);
    const int totalWaves    = gridDim.x * wavesPerBlock;

    // Wave-uniform grid-stride loop: EXEC is all-ones at every WMMA.
    for (int tile = waveId; tile < numTiles; tile += totalWaves) {
        int r  = tile * 16 + row;
        int rc = (r < n) ? r : (n - 1);                 // branch-free clamp
        const float* xp = x + (size_t)rc * 6;

        // B operand loads: exactly 48 float2 per 16-row tile (96 floats),
        // every fetched byte is used once.
        v2f bv0 = *(const v2f*)(xp + hi * 2);           // k={0,1} / k={2,3}
        v2f bv1 = {0.f, 0.f};
        if (hi == 0)                                    // k={4,5}; k={6,7}=0
            bv1 = *(const v2f*)(xp + 4);

        // Prefetch this wave's next tile while we compute.
        int rn = r + totalWaves * 16;
        if (rn < n)
            __builtin_prefetch(x + (size_t)rn * 6, 0, 1);

        // ---- Layer 1: two chained f32 WMMAs (K = 4 + 4-with-padding) ----
        v8f c = {};
        c = __builtin_amdgcn_wmma_f32_16x16x4_f32(
                false, a0, false, bv0, (short)0, c, false, false);
        c = __builtin_amdgcn_wmma_f32_16x16x4_f32(
                false, a1, false, bv1, (short)0, c, false, false);

        // Lane n (hi==0): c[m] = <W1[m,:], x[r,:]>  for m = 0..7
        float h[8];
#pragma unroll
        for (int j = 0; j < 8; ++j)
            h[j] = fmaxf(c[j] + B1r[j], 0.f);

        // ---- Layer 7: 8 -> 4, relu ----
        float a7[4];
#pragma unroll
        for (int i = 0; i < 4; ++i) {
            float s = B7r[i];
#pragma unroll
            for (int j = 0; j < 8; ++j) s = fmaf(W7r[i][j], h[j], s);
            a7[i] = fmaxf(s, 0.f);
        }

        // ---- Layer 8: 4 -> 2, relu ----
        float a8[2];
#pragma unroll
        for (int i = 0; i < 2; ++i) {
            float s = B8r[i];
#pragma unroll
            for (int j = 0; j < 4; ++j) s = fmaf(W8r[i][j], a7[j], s);
            a8[i] = fmaxf(s, 0.f);
        }

        // ---- Layer 9: 2 -> 1, then numerically stable softplus ----
        float z  = fmaf(W9r[0], a8[0], fmaf(W9r[1], a8[1], B9v));
        float sp = fmaxf(z, 0.f) + log1pf(expf(-fabsf(z)));

        if (hi == 0 && r < n)
            out[r] = sp;
    }
}

extern "C" void kernel_launch(void* const* d_in, const int* in_sizes, int n_in,
                              void* d_out, int out_size, void* d_ws, size_t ws_size,
                              hipStream_t stream) {
    (void)n_in; (void)out_size; (void)d_ws; (void)ws_size;
    const float* x  = (const float*)d_in[0];
    const float* W1 = (const float*)d_in[1];
    const float* b1 = (const float*)d_in[2];
    const float* W7 = (const float*)d_in[3];
    const float* b7 = (const float*)d_in[4];
    const float* W8 = (const float*)d_in[5];
    const float* b8 = (const float*)d_in[6];
    const float* W9 = (const float*)d_in[7];
    const float* b9 = (const float*)d_in[8];
    float* out = (float*)d_out;

    const int n        = in_sizes[0] / 6;       // rows
    const int numTiles = (n + 15) / 16;
    const int wavesPerBlock = 256 / 32;
    int blocks = (numTiles + wavesPerBlock - 1) / wavesPerBlock;
    if (blocks > 4096) blocks = 4096;           // ~32K wave32 in flight
    if (blocks < 1)    blocks = 1;

    mlp6_8_4_2_1_softplus<<<blocks, 256, 0, stream>>>(
        x, W1, b1, W7, b7, W8, b8, W9, b9, out, n);
}